// multi_scale_initdata_23244363006225
// MI455X (gfx1250) — compile-verified
//
#include <hip/hip_runtime.h>

// ---------------------------------------------------------------------------
// Multi-scale avg/max pooling + blend, MI455X (gfx1250).
// x: (8, 4096, 3, 256) f32, w: (3584) f32 -> out: (8, 3584, 3, 256) f32
// Pure streaming kernel (~189 MB total traffic -> ~8.1 us at 23.3 TB/s).
// Strategy: read input ONCE, compute pool2/4/8 hierarchically, use the CDNA5
// async global->LDS copy path with double buffering per wave (ASYNCcnt).
// ---------------------------------------------------------------------------

typedef float v4f __attribute__((ext_vector_type(4)));
// Matches the builtin's parameter type from the round-1 diagnostic:
// '__attribute__((__vector_size__(4 * sizeof(int)))) int __device__ *'
typedef int v4i __attribute__((vector_size(16)));
typedef __attribute__((address_space(1))) v4i* gptr_t;  // global int4*
typedef __attribute__((address_space(3))) v4i* lptr_t;  // LDS int4*

#define B_N 8
#define T_N 4096
#define C_N 3
#define D_N 256
#define MAXLEN 3584
#define ROWSTRIDE (C_N * D_N)            // 768 floats between consecutive t rows
#define N8 (T_N / 8)                     // 512 groups of 8 rows per (b,c)
#define GROUPS (B_N * C_N * N8)          // 12288
#define CHAIN 4                          // groups per wave (512 % 4 == 0)
#define WPB 4                            // waves per block
#define THREADS (WPB * 32)
#define NBLOCKS (GROUPS / (CHAIN * WPB)) // 768

#if defined(__AMDGCN__) && __has_builtin(__builtin_amdgcn_global_load_async_to_lds_b128)
#define USE_ASYNC 1
#else
#define USE_ASYNC 0
#endif

static __device__ __forceinline__ v4f vmax4(v4f a, v4f b) {
  v4f r;
  r.x = fmaxf(a.x, b.x);
  r.y = fmaxf(a.y, b.y);
  r.z = fmaxf(a.z, b.z);
  r.w = fmaxf(a.w, b.w);
  return r;
}

// From 8 consecutive-t float4 values produce the 7 pooled outputs:
// avg[0..3]/mx[0..3] = pool2, avg[4..5]/mx[4..5] = pool4, avg[6]/mx[6] = pool8
static __device__ __forceinline__ void reduce7(const v4f a[8], v4f avg[7], v4f mx[7]) {
  v4f s2_0 = a[0] + a[1], s2_1 = a[2] + a[3], s2_2 = a[4] + a[5], s2_3 = a[6] + a[7];
  v4f m2_0 = vmax4(a[0], a[1]), m2_1 = vmax4(a[2], a[3]);
  v4f m2_2 = vmax4(a[4], a[5]), m2_3 = vmax4(a[6], a[7]);
  v4f s4_0 = s2_0 + s2_1, s4_1 = s2_2 + s2_3;
  v4f m4_0 = vmax4(m2_0, m2_1), m4_1 = vmax4(m2_2, m2_3);
  v4f s8 = s4_0 + s4_1;
  v4f m8 = vmax4(m4_0, m4_1);
  avg[0] = s2_0 * 0.5f;   avg[1] = s2_1 * 0.5f;
  avg[2] = s2_2 * 0.5f;   avg[3] = s2_3 * 0.5f;
  avg[4] = s4_0 * 0.25f;  avg[5] = s4_1 * 0.25f;
  avg[6] = s8 * 0.125f;
  mx[0] = m2_0; mx[1] = m2_1; mx[2] = m2_2; mx[3] = m2_3;
  mx[4] = m4_0; mx[5] = m4_1; mx[6] = m8;
}

#if USE_ASYNC
// Issue one group's worth of async copies: 8 rows x 1KB, 16 x B128 issues
// (each issue: 32 lanes x 16B = 512B contiguous).
static __device__ __forceinline__ void issue_group(const float* __restrict__ src0,
                                                   float* ldsbuf, int lane) {
#pragma unroll
  for (int r = 0; r < 8; ++r) {
    const float* s = src0 + r * ROWSTRIDE;
    float* d = ldsbuf + r * D_N;
    __builtin_amdgcn_global_load_async_to_lds_b128(
        (gptr_t)(s + lane * 4), (lptr_t)(d + lane * 4), 0, 0);
    __builtin_amdgcn_global_load_async_to_lds_b128(
        (gptr_t)(s + 128 + lane * 4), (lptr_t)(d + 128 + lane * 4), 0, 0);
  }
}
#endif

__global__ __launch_bounds__(THREADS) void msp_kernel(const float* __restrict__ x,
                                                      const float* __restrict__ w,
                                                      float* __restrict__ out) {
#if USE_ASYNC
  __shared__ alignas(16) float lds[WPB][2][8 * D_N];  // 4 waves * 2 * 8KB = 64KB
#endif
  const int lane      = (int)(threadIdx.x & 31u);
  const int waveInBlk = (int)(threadIdx.x >> 5);
  const int waveId    = (int)blockIdx.x * WPB + waveInBlk;
  const int g0        = waveId * CHAIN;

  // Chain stays inside one (b,c) since N8 % CHAIN == 0.
  const int n8_0 = g0 % N8;
  const int bc   = g0 / N8;
  const int c    = bc % C_N;
  const int b    = bc / C_N;

  const size_t inBase  = (size_t)b * (size_t)T_N * ROWSTRIDE + (size_t)c * D_N;
  const size_t outBase = (size_t)b * (size_t)MAXLEN * ROWSTRIDE + (size_t)c * D_N;

  // Lane owns two separated d-quads so every LDS/global access is a
  // contiguous 512B slice across the wave (conflict-free, fully coalesced).
  const int d0 = lane * 4;
  const int d1 = 128 + lane * 4;

#if USE_ASYNC
  issue_group(x + inBase + (size_t)(n8_0 * 8) * ROWSTRIDE, &lds[waveInBlk][0][0], lane);
#endif

#pragma unroll
  for (int j = 0; j < CHAIN; ++j) {
    const int n8 = n8_0 + j;
    v4f alo[8], ahi[8];

#if USE_ASYNC
    if (j + 1 < CHAIN) {
      // Make sure our own ds_loads of the buffer we are about to overwrite
      // have fully completed before the async engine writes into it.
      asm volatile("s_wait_dscnt 0" ::: "memory");
      issue_group(x + inBase + (size_t)((n8 + 1) * 8) * ROWSTRIDE,
                  &lds[waveInBlk][(j + 1) & 1][0], lane);
      // 16 issues outstanding for the next group; group j is landed when
      // ASYNCcnt drops to <= 16 (async loads complete in order).
      asm volatile("s_wait_asynccnt 16" ::: "memory");
    } else {
      asm volatile("s_wait_asynccnt 0" ::: "memory");
    }
    const float* rowsrc = &lds[waveInBlk][j & 1][0];
#pragma unroll
    for (int r = 0; r < 8; ++r) {
      alo[r] = *(const v4f*)(rowsrc + r * D_N + d0);
      ahi[r] = *(const v4f*)(rowsrc + r * D_N + d1);
    }
#else
#pragma unroll
    for (int r = 0; r < 8; ++r) {
      const float* s = x + inBase + (size_t)(n8 * 8 + r) * ROWSTRIDE;
      alo[r] = *(const v4f*)(s + d0);
      ahi[r] = *(const v4f*)(s + d1);
    }
#endif

    v4f slo[7], mlo[7], shi[7], mhi[7];
    reduce7(alo, slo, mlo);
    reduce7(ahi, shi, mhi);

    // Output positions: pool2 segment [0,2048), pool4 [2048,3072), pool8 [3072,3584)
    const int ls[7] = {n8 * 4 + 0, n8 * 4 + 1, n8 * 4 + 2, n8 * 4 + 3,
                       2048 + n8 * 2 + 0, 2048 + n8 * 2 + 1, 3072 + n8};
#pragma unroll
    for (int q = 0; q < 7; ++q) {
      const int l = ls[q];
      const float wv = w[l];                 // uniform per wave (broadcast load)
      const float iv = 1.0f - wv;
      float* o = out + outBase + (size_t)l * ROWSTRIDE;
      v4f rlo = slo[q] * wv + mlo[q] * iv;
      v4f rhi = shi[q] * wv + mhi[q] * iv;
      *(v4f*)(o + d0) = rlo;
      *(v4f*)(o + d1) = rhi;
    }
  }
}

extern "C" void kernel_launch(void* const* d_in, const int* in_sizes, int n_in,
                              void* d_out, int out_size, void* d_ws, size_t ws_size,
                              hipStream_t stream) {
  (void)in_sizes; (void)n_in; (void)out_size; (void)d_ws; (void)ws_size;
  const float* x = (const float*)d_in[0];
  const float* w = (const float*)d_in[1];
  float* out = (float*)d_out;
  msp_kernel<<<NBLOCKS, THREADS, 0, stream>>>(x, w, out);
}